// TrustWI_5145370821142
// MI455X (gfx1250) — compile-verified
//
#include <hip/hip_runtime.h>
#include <math.h>

#define NROWS  32768
#define KCLS   40
#define MT     64      // rows per block
#define NCH    128     // HID columns per chunk
#define KST    32      // k staging depth
#define TPB    256     // 8 waves of 32
#define XST    36      // padded LDS stride for x tile (bank-conflict free, 16B-aligned)
#define HST    129     // padded LDS stride for h buffer
#define BN_EPS 1e-5f

typedef float v2f __attribute__((ext_vector_type(2)));
typedef float v8f __attribute__((ext_vector_type(8)));
typedef int   v4i __attribute__((vector_size(16)));   // matches builtin param type

#ifndef __has_builtin
#define __has_builtin(x) 0
#endif

// ---- gfx1250 async global->LDS copy (ASYNCcnt-tracked, bypasses VGPRs) ----
__device__ __forceinline__ void async_b128(const void* g, void* l)
{
#if __has_builtin(__builtin_amdgcn_global_load_async_to_lds_b128)
    typedef __attribute__((address_space(1))) v4i* gp_t;
    typedef __attribute__((address_space(3))) v4i* lp_t;
    __builtin_amdgcn_global_load_async_to_lds_b128(
        (gp_t)(void*)(uintptr_t)g, (lp_t)l, 0, 0);
#else
    // VDST = LDS byte offset (flat LDS addr truncates to addr[31:0] per ISA 10.2)
    const unsigned loff = (unsigned)(uintptr_t)l;
    asm volatile("global_load_async_to_lds_b128 %0, %1, off"
                 :: "v"(loff), "v"((unsigned long long)(uintptr_t)g)
                 : "memory");
#endif
}

#if __has_builtin(__builtin_amdgcn_s_wait_asynccnt)
#define WAIT_ASYNCCNT(n) __builtin_amdgcn_s_wait_asynccnt(n)
#else
#define WAIT_ASYNCCNT(n) asm volatile("s_wait_asynccnt %0" :: "i"(n) : "memory")
#endif

__global__ __launch_bounds__(TPB) void fused_mlp_kernel(
    const float* __restrict__ x,   const float* __restrict__ xg,
    const float* __restrict__ w1,  const float* __restrict__ b1,
    const float* __restrict__ g1,  const float* __restrict__ be1,
    const float* __restrict__ rm1, const float* __restrict__ rv1,
    const float* __restrict__ w2,  const float* __restrict__ b2,
    const float* __restrict__ wp1, const float* __restrict__ bp1,
    const float* __restrict__ gp1, const float* __restrict__ bep1,
    const float* __restrict__ rmp1,const float* __restrict__ rvp1,
    const float* __restrict__ wp2, const float* __restrict__ bp2,
    float* __restrict__ out)
{
    const int enc  = blockIdx.y;            // 0: x, 1: x_glo, 2: concat(x, x_glo)
    const int row0 = blockIdx.x * MT;
    const int tid  = threadIdx.x;
    const int lane = tid & 31;
    const int wid  = tid >> 5;
    const int mr   = wid >> 1;              // 0..3 : 16-row group
    const int nc   = wid & 1;               // 0..1 : 64-col group

    const float *W1, *B1, *G1, *BE, *RM, *RV, *W2, *B2;
    int KTOT, HID;
    if (enc < 2) { W1=w1;  B1=b1;  G1=g1;  BE=be1;  RM=rm1;  RV=rv1;  W2=w2;  B2=b2;  KTOT=1024; HID=1024; }
    else         { W1=wp1; B1=bp1; G1=gp1; BE=bep1; RM=rmp1; RV=rvp1; W2=wp2; B2=bp2; KTOT=2048; HID=2048; }

    extern __shared__ float smem[];
    float* xs   = smem;                      // 2 * MT*XST  = 4608 floats (double buffered)
    float* wsa  = xs   + 2 * MT * XST;       // 2 * KST*NCH = 8192 floats (double buffered)
    float* hbuf = wsa  + 2 * KST * NCH;      // MT*HST      = 8256 floats
    float* w2s  = hbuf + MT * HST;           // NCH*KCLS    = 5120 floats

    // second-GEMM ownership: thread -> (row r2, 10 output classes starting at cg*10)
    const int r2 = tid & 63;
    const int cg = tid >> 6;                 // 0..3
    float acc2[10];
#pragma unroll
    for (int i = 0; i < 10; ++i) acc2[i] = 0.f;

    const int koff = (lane >> 4) << 1;       // 0 for lanes 0-15, 2 for lanes 16-31
    const int ml   = lane & 15;
    const int mh   = (lane >> 4) << 3;       // 0 or 8 (C/D row offset)

    // issues 6 async b128 per thread: x tile (2) + w1 tile (4)
    auto issue_stage = [&](int k0, int bufi, int n0) {
        float* xsb = xs  + bufi * (MT * XST);
        float* wsb = wsa + bufi * (KST * NCH);
        const float* src;
        int kc;
        if (enc == 0)      { src = x;  kc = k0; }
        else if (enc == 1) { src = xg; kc = k0; }
        else               { src = (k0 < 1024) ? x : xg; kc = k0 & 1023; }
#pragma unroll
        for (int i = 0; i < 2; ++i) {
            const int f4 = tid + i * TPB;          // 0..511
            const int r  = f4 >> 3;                // 8 float4 per row
            const int c4 = (f4 & 7) << 2;
            async_b128(&src[(size_t)(row0 + r) * 1024 + kc + c4], &xsb[r * XST + c4]);
        }
#pragma unroll
        for (int i = 0; i < 4; ++i) {
            const int f4 = tid + i * TPB;          // 0..1023
            const int r  = f4 >> 5;                // 32 float4 per row
            const int c4 = (f4 & 31) << 2;
            async_b128(&W1[(size_t)(k0 + r) * HID + n0 + c4], &wsb[r * NCH + c4]);
        }
    };

    const int nchunks = HID / NCH;
    const int SK = KTOT / KST;
    for (int nci = 0; nci < nchunks; ++nci) {
        const int n0 = nci * NCH;

        v8f acc[4];
#pragma unroll
        for (int t = 0; t < 4; ++t)
#pragma unroll
            for (int g = 0; g < 8; ++g) acc[t][g] = 0.f;

        // software pipeline: stage s+1 in flight while computing stage s
        issue_stage(0, 0, n0);
        for (int s = 0; s < SK; ++s) {
            const int buf = s & 1;
            __syncthreads();                       // prior readers of buf^1 are done
            if (s + 1 < SK) {
                issue_stage((s + 1) * KST, buf ^ 1, n0);
                WAIT_ASYNCCNT(6);                  // in-order: retires stage s's 6 loads
            } else {
                WAIT_ASYNCCNT(0);
            }
            __syncthreads();                       // stage s visible to all waves

            const float* xsb = xs  + buf * (MT * XST);
            const float* wsb = wsa + buf * (KST * NCH);
            // ---- fp32 WMMA: 8 k-steps of 4 ----
#pragma unroll
            for (int kk = 0; kk < KST; kk += 4) {
                v2f a;
                a.x = xsb[(mr * 16 + ml) * XST + kk + koff];
                a.y = xsb[(mr * 16 + ml) * XST + kk + koff + 1];
#pragma unroll
                for (int t = 0; t < 4; ++t) {
                    const int n = nc * 64 + t * 16 + ml;
                    v2f b;
                    b.x = wsb[(kk + koff)     * NCH + n];
                    b.y = wsb[(kk + koff + 1) * NCH + n];
                    acc[t] = __builtin_amdgcn_wmma_f32_16x16x4_f32(
                        false, a, false, b, (short)0, acc[t], false, false);
                }
            }
        }

        // ---- epilogue: bias + BN(eval) + ReLU -> hbuf ----
#pragma unroll
        for (int t = 0; t < 4; ++t) {
            const int nl = nc * 64 + t * 16 + ml;
            const int j  = n0 + nl;
            const float sc = G1[j] * rsqrtf(RV[j] + BN_EPS);
            const float sh = (B1[j] - RM[j]) * sc + BE[j];
#pragma unroll
            for (int g = 0; g < 8; ++g) {
                float v = acc[t][g] * sc + sh;     // (acc + b1 - rm)*rsqrt(rv+eps)*g + beta
                v = fmaxf(v, 0.f);
                hbuf[(mr * 16 + mh + g) * HST + nl] = v;
            }
        }
        // ---- stage w2 chunk (contiguous 5120 floats) via async copies ----
#pragma unroll
        for (int i = 0; i < 5; ++i) {
            const int f4 = tid + i * TPB;          // 0..1279
            async_b128(&W2[(size_t)n0 * KCLS + f4 * 4], &w2s[f4 * 4]);
        }
        WAIT_ASYNCCNT(0);
        __syncthreads();

        // ---- small GEMM2 accumulate: out2[r2, cg*10 .. +9] += h_chunk @ w2_chunk ----
        {
            const float2* w2v  = reinterpret_cast<const float2*>(w2s);
            const int cbase    = cg * 5;           // float2 column base
            const int hrow     = r2 * HST;
#pragma unroll 4
            for (int kk = 0; kk < NCH; ++kk) {
                const float hv = hbuf[hrow + kk];
#pragma unroll
                for (int i = 0; i < 5; ++i) {
                    const float2 w = w2v[kk * (KCLS / 2) + cbase + i];
                    acc2[2 * i + 0] += hv * w.x;
                    acc2[2 * i + 1] += hv * w.y;
                }
            }
        }
    }

    // ---- bias2 + softplus + 1 -> alpha_{enc+1} ----
    {
        float* dst = out + (size_t)enc * NROWS * KCLS
                         + (size_t)(row0 + r2) * KCLS + cg * 10;
#pragma unroll
        for (int i = 0; i < 10; ++i) {
            const float v  = acc2[i] + B2[cg * 10 + i];
            const float sp = (v > 20.f) ? v : log1pf(__expf(v));
            dst[i] = sp + 1.f;
        }
    }
}

// ---------------- Dempster-Shafer combination ----------------

__device__ __forceinline__ void ds_comb(const float* a, const float* b, float* o)
{
    float S1 = 0.f, S2 = 0.f;
#pragma unroll
    for (int k = 0; k < KCLS; ++k) { S1 += a[k]; S2 += b[k]; }
    const float i1 = 1.f / S1, i2 = 1.f / S2;
    const float u1 = KCLS * i1, u2 = KCLS * i2;
    float b1v[KCLS], b2v[KCLS];
    float sb1 = 0.f, sb2 = 0.f, sbb = 0.f;
#pragma unroll
    for (int k = 0; k < KCLS; ++k) {
        b1v[k] = (a[k] - 1.f) * i1;
        b2v[k] = (b[k] - 1.f) * i2;
        sb1 += b1v[k]; sb2 += b2v[k]; sbb += b1v[k] * b2v[k];
    }
    const float C    = sb1 * sb2 - sbb;
    const float invd = 1.f / (1.f - C);
    const float u_a  = u1 * u2 * invd;
    const float S_a  = (float)KCLS / u_a;
#pragma unroll
    for (int k = 0; k < KCLS; ++k) {
        const float b_a = (b1v[k] * b2v[k] + b1v[k] * u2 + b2v[k] * u1) * invd;
        o[k] = b_a * S_a + 1.f;
    }
}

__global__ __launch_bounds__(256) void combine_kernel(float* __restrict__ out)
{
    const int row = blockIdx.x * 256 + threadIdx.x;
    if (row >= NROWS) return;
    const size_t NK = (size_t)NROWS * KCLS;
    float a1[KCLS], a2[KCLS], a3[KCLS], t[KCLS], r[KCLS];
    const float* p1 = out + (size_t)row * KCLS;
    const float* p2 = out + NK + (size_t)row * KCLS;
    const float* p3 = out + 2 * NK + (size_t)row * KCLS;
#pragma unroll
    for (int k = 0; k < KCLS; ++k) { a1[k] = p1[k]; a2[k] = p2[k]; a3[k] = p3[k]; }
    ds_comb(a1, a2, t);
    ds_comb(t, a3, r);
    float* po = out + 3 * NK + (size_t)row * KCLS;
#pragma unroll
    for (int k = 0; k < KCLS; ++k) po[k] = r[k];
}

extern "C" void kernel_launch(void* const* d_in, const int* in_sizes, int n_in,
                              void* d_out, int out_size, void* d_ws, size_t ws_size,
                              hipStream_t stream)
{
    (void)in_sizes; (void)n_in; (void)out_size; (void)d_ws; (void)ws_size;
    const float* x    = (const float*)d_in[0];
    const float* xg   = (const float*)d_in[1];
    const float* w1   = (const float*)d_in[2];
    const float* b1   = (const float*)d_in[3];
    const float* g1   = (const float*)d_in[4];
    const float* be1  = (const float*)d_in[5];
    const float* rm1  = (const float*)d_in[6];
    const float* rv1  = (const float*)d_in[7];
    const float* w2   = (const float*)d_in[8];
    const float* b2   = (const float*)d_in[9];
    const float* wp1  = (const float*)d_in[10];
    const float* bp1  = (const float*)d_in[11];
    const float* gp1  = (const float*)d_in[12];
    const float* bep1 = (const float*)d_in[13];
    const float* rmp1 = (const float*)d_in[14];
    const float* rvp1 = (const float*)d_in[15];
    const float* wp2  = (const float*)d_in[16];
    const float* bp2  = (const float*)d_in[17];
    float* out = (float*)d_out;

    const size_t smem_bytes =
        (size_t)(2 * MT * XST + 2 * KST * NCH + MT * HST + NCH * KCLS) * sizeof(float);

    dim3 grid(NROWS / MT, 3);
    fused_mlp_kernel<<<grid, TPB, smem_bytes, stream>>>(
        x, xg, w1, b1, g1, be1, rm1, rv1, w2, b2,
        wp1, bp1, gp1, bep1, rmp1, rvp1, wp2, bp2, out);

    combine_kernel<<<(NROWS + 255) / 256, 256, 0, stream>>>(out);
}